// SynthLocalLoss_MDN_8589934592313
// MI455X (gfx1250) — compile-verified
//
#include <hip/hip_runtime.h>
#include <math.h>

// Problem constants (from reference)
#define NR     4096
#define NLID   16384
#define KMIX   8
#define TOPK   10
#define CFEAT  16
#define LOG2PI_F 1.8378770664093453f

// batch folded into 4th coordinate: mismatch penalty >= 512^2 = 262,144
// max in-batch d2 = 40^2 + 175^2 + 175^2 = 62,850  ->  threshold 1.5e5
// scale 512 keeps every WMMA intermediate < 2^24 -> integer-exact f32
#define BATCH_SCALE   512.0f
#define MATCH_THRESH  1.5e5f
#define FINF          3.0e38f

#define WAVES_PER_BLOCK 8
#define NTHREADS (WAVES_PER_BLOCK * 32)     // 256
#define CHUNK  (NLID / WAVES_PER_BLOCK)     // 2048 lidar pts per wave
#define TILES  (CHUNK / 16)                 // 128 WMMA tiles per wave
#define NSL    (TOPK + 1)                   // padded list length (cursor-safe)
#define NSRC   (WAVES_PER_BLOCK * 2)        // 16 partial lists per radar row

typedef __attribute__((ext_vector_type(2))) float v2f;
typedef __attribute__((ext_vector_type(8))) float v8f;

// ---------------------------------------------------------------------------
// Phase 0: zero the accumulators (occ_sum, cnt, logp_sum, int_sum)
// ---------------------------------------------------------------------------
__global__ void init_acc_kernel(float* __restrict__ acc) {
    if (threadIdx.x < 4) acc[threadIdx.x] = 0.0f;
}

// ---------------------------------------------------------------------------
// Phase 1: batched masked top-10 nearest neighbor via two chained
// V_WMMA_F32_16X16X4_F32 (homogeneous-coordinate distance):
//   D1 = lidar(16x4) x (-2*radar)(4x16)              (cross terms)
//   D2 = (|l|^2,1,0,0)(16x4) x (1,|r|^2,0,0)(4x16) + D1  = full d^2
// Per-lane top-10 lists live in LDS slots (register-promoted by LLVM); the
// scan loop carries only `thr` + 2-deep load pipeline.
// ---------------------------------------------------------------------------
__global__ __launch_bounds__(NTHREADS) void match_topk_kernel(
    const int* __restrict__ radar_idx,   // (NR,4)  [batch,d1,d2,d3]
    const int* __restrict__ lidar_idx,   // (NL,4)
    int* __restrict__ matched_out,       // (NR)
    int* __restrict__ nn_out)            // (NR,TOPK)
{
    __shared__ float stopv[NTHREADS][NSL];
    __shared__ int   stixv[NTHREADS][NSL];

    const int tid  = threadIdx.x;
    const int lane = tid & 31;
    const int wave = tid >> 5;
    const int rblk = blockIdx.x * 16;
    const bool lo  = (lane < 16);

    // private sorted list (ascending) in LDS, entry TOPK is permanent +INF pad
    #pragma unroll
    for (int s = 0; s < NSL; ++s) { stopv[tid][s] = FINF; stixv[tid][s] = 0; }
    float thr = FINF;   // current 10th-best (only loop-carried scan state)

    // ---- B operands: radar point (lane&15), fixed per wave (branchless) ----
    const int4 ri = ((const int4*)radar_idx)[rblk + (lane & 15)];
    const float r0 = (float)ri.y, r1 = (float)ri.z, r2c = (float)ri.w;
    const float r3 = BATCH_SCALE * (float)ri.x;
    const float r2sum = r0*r0 + r1*r1 + r2c*r2c + r3*r3;
    v2f b1, b2;
    b1.x = -2.0f * (lo ? r0 : r2c);
    b1.y = -2.0f * (lo ? r1 : r3);
    b2.x = lo ? 1.0f  : 0.0f;
    b2.y = lo ? r2sum : 0.0f;
    const float a2y = lo ? 1.0f : 0.0f;

    const int base0 = wave * CHUNK;
    const int mofs  = lo ? 0 : 8;            // which 8 D-rows this lane holds
    const int lsub  = lane & 15;

    // 2-deep software pipeline of lidar tile loads (two b128 in flight)
    int4 li  = ((const int4*)lidar_idx)[base0 + lsub];
    int4 li1 = ((const int4*)lidar_idx)[base0 + 16 + lsub];
    for (int t = 0; t < TILES; ++t) {
        const int nt = (t + 2 < TILES) ? (t + 2) : (TILES - 1);
        const int4 nli = ((const int4*)lidar_idx)[base0 + nt * 16 + lsub];

        const float l0 = (float)li.y, l1 = (float)li.z, l2 = (float)li.w;
        const float l3 = BATCH_SCALE * (float)li.x;
        const float l2sum = l0*l0 + l1*l1 + l2*l2 + l3*l3;
        v2f a1, a2;
        a1.x = lo ? l0 : l2;
        a1.y = lo ? l1 : l3;
        a2.x = lo ? l2sum : 0.0f;
        a2.y = a2y;

        v8f c = {};
        v8f d = __builtin_amdgcn_wmma_f32_16x16x4_f32(
            false, a1, false, b1, (short)0, c, false, false);
        d = __builtin_amdgcn_wmma_f32_16x16x4_f32(
            false, a2, false, b2, (short)0, d, false, false);

        // tile-level prefilter: one compare in the common path
        const float m01 = fminf(d[0], d[1]), m23 = fminf(d[2], d[3]);
        const float m45 = fminf(d[4], d[5]), m67 = fminf(d[6], d[7]);
        const float tmin = fminf(fminf(m01, m23), fminf(m45, m67));

        if (tmin < thr) {                      // cold block, execz-skipped
            const int lb = base0 + t * 16 + mofs;
            #pragma unroll
            for (int j = 0; j < 8; ++j) {
                float v = d[j];
                if (v < thr) {
                    int vi = lb + j;
                    #pragma unroll
                    for (int s = 0; s < TOPK; ++s) {
                        const float cur = stopv[tid][s];
                        const int   ci  = stixv[tid][s];
                        const bool  lt  = v < cur;
                        const float nv  = lt ? v  : cur;
                        const int   ni  = lt ? vi : ci;
                        stopv[tid][s] = nv;  stixv[tid][s] = ni;
                        v  = lt ? cur : v;
                        vi = lt ? ci  : vi;
                        if (s == TOPK - 1) thr = nv;   // new 10th-best
                    }
                }
            }
        }
        li = li1; li1 = nli;
    }
    __syncthreads();

    // ---- 16-way cursor merge of sorted lists (8 waves x 2 half-lanes) ----
    if (tid < 16) {
        const int r = rblk + tid;
        int srcs[NSRC]; int pos[NSRC];
        #pragma unroll
        for (int w = 0; w < NSRC; ++w) {
            srcs[w] = (w >> 1) * 32 + ((w & 1) << 4) + tid;  // wave, half, lane
            pos[w]  = 0;
        }
        float best0 = FINF;
        for (int p = 0; p < TOPK; ++p) {
            float mv = FINF; int mi = 0, mw = 0;
            #pragma unroll
            for (int w = 0; w < NSRC; ++w) {
                const float vv = stopv[srcs[w]][pos[w]];
                const int   ii = stixv[srcs[w]][pos[w]];
                const bool  lt = vv < mv;
                mv = lt ? vv : mv;
                mi = lt ? ii : mi;
                mw = lt ? w  : mw;
            }
            #pragma unroll
            for (int w = 0; w < NSRC; ++w) pos[w] += (mw == w) ? 1 : 0;
            nn_out[r * TOPK + p] = mi;
            if (p == 0) best0 = mv;
        }
        matched_out[r] = (best0 < MATCH_THRESH) ? 1 : 0;
    }
}

// ---------------------------------------------------------------------------
// Phase 2: per-(radar, t) loss terms; wave shfl-reduction + LDS cross-wave
// reduction, then ONE atomicAdd per accumulator per block (256x fewer RMWs).
//   acc[0]=sum occ softplus, acc[1]=matched count,
//   acc[2]=sum masked logp,  acc[3]=sum masked int term
// ---------------------------------------------------------------------------
__global__ __launch_bounds__(256) void loss_kernel(
    const float* __restrict__ mu_off,    // (NR,K,3)
    const float* __restrict__ log_sig,   // (NR,K,3)
    const float* __restrict__ mu_int,    // (NR,K)
    const float* __restrict__ occ_logit, // (NR,K)
    const float* __restrict__ mix_logit, // (NR,K)
    const float* __restrict__ lidar_feat,// (NL,C)
    const int*   __restrict__ radar_idx, // (NR,4)
    const int*   __restrict__ lidar_idx, // (NL,4)
    const int*   __restrict__ matched,   // (NR)
    const int*   __restrict__ nn,        // (NR,TOPK)
    float* __restrict__ acc)
{
    __shared__ float part[8][4];

    const int tid = blockIdx.x * blockDim.x + threadIdx.x;
    float c0 = 0.0f, c1 = 0.0f, c2 = 0.0f, c3 = 0.0f;

    if (tid < NR * TOPK) {
        const int r = tid / TOPK;
        const int t = tid - r * TOPK;
        const int m = matched[r];

        if (t == 0) {
            float oa = -1.0e30f;
            #pragma unroll
            for (int k = 0; k < KMIX; ++k) oa = fmaxf(oa, occ_logit[r * KMIX + k]);
            const float z = m ? -oa : oa;                    // softplus(+/-occ_any)
            c0 = fmaxf(z, 0.0f) + log1pf(expf(-fabsf(z)));
            c1 = (float)m;
        }
        if (m) {
            const int j = nn[r * TOPK + t];
            const int4 ri = ((const int4*)radar_idx)[r];
            const int4 li = ((const int4*)lidar_idx)[j];
            // y = offs_zyx[..., ::-1]  ->  (d3, d2, d1) diffs
            const float yv[3] = { (float)(li.w - ri.w),
                                  (float)(li.z - ri.z),
                                  (float)(li.y - ri.y) };

            // log_softmax of mixture logits
            float ml[KMIX]; float mx = -1.0e30f;
            #pragma unroll
            for (int k = 0; k < KMIX; ++k) { ml[k] = mix_logit[r * KMIX + k]; mx = fmaxf(mx, ml[k]); }
            float se = 0.0f;
            #pragma unroll
            for (int k = 0; k < KMIX; ++k) se += expf(ml[k] - mx);
            const float lse = logf(se) + mx;

            // component log posteriors
            float lp[KMIX]; float lmax = -1.0e30f;
            #pragma unroll
            for (int k = 0; k < KMIX; ++k) {
                const float* mu = &mu_off[(r * KMIX + k) * 3];
                const float* ls = &log_sig[(r * KMIX + k) * 3];
                float s = 0.0f;
                #pragma unroll
                for (int d = 0; d < 3; ++d) {
                    const float lsd  = ls[d];
                    const float sig2 = expf(2.0f * lsd) + 1e-12f;
                    const float diff = yv[d] - mu[d];
                    s += diff * diff / sig2 + 2.0f * lsd + LOG2PI_F;
                }
                lp[k] = -0.5f * s + (ml[k] - lse);
                lmax  = fmaxf(lmax, lp[k]);
            }
            float s2 = 0.0f;
            #pragma unroll
            for (int k = 0; k < KMIX; ++k) s2 += expf(lp[k] - lmax);
            const float logp = logf(s2) + lmax;
            c2 = logp;

            // gt_int = mean of feature columns 3,7,11,15
            const float gt = 0.25f * (lidar_feat[j * CFEAT + 3]  + lidar_feat[j * CFEAT + 7] +
                                      lidar_feat[j * CFEAT + 11] + lidar_feat[j * CFEAT + 15]);
            float term = 0.0f;
            #pragma unroll
            for (int k = 0; k < KMIX; ++k)
                term += expf(lp[k] - logp) * fabsf(mu_int[r * KMIX + k] - gt);
            c3 = term;
        }
    }

    // ---- wave-level reduction (wave32) ----
    #pragma unroll
    for (int off = 16; off > 0; off >>= 1) {
        c0 += __shfl_down(c0, off, 32);
        c1 += __shfl_down(c1, off, 32);
        c2 += __shfl_down(c2, off, 32);
        c3 += __shfl_down(c3, off, 32);
    }
    const int lane = threadIdx.x & 31;
    const int wid  = threadIdx.x >> 5;
    if (lane == 0) {
        part[wid][0] = c0; part[wid][1] = c1;
        part[wid][2] = c2; part[wid][3] = c3;
    }
    __syncthreads();
    if (threadIdx.x == 0) {
        float s0 = 0.0f, s1 = 0.0f, s2 = 0.0f, s3 = 0.0f;
        #pragma unroll
        for (int w = 0; w < 8; ++w) {
            s0 += part[w][0]; s1 += part[w][1];
            s2 += part[w][2]; s3 += part[w][3];
        }
        atomicAdd(&acc[0], s0);
        atomicAdd(&acc[1], s1);
        atomicAdd(&acc[2], s2);
        atomicAdd(&acc[3], s3);
    }
}

// ---------------------------------------------------------------------------
// Phase 3: combine into the scalar loss
// ---------------------------------------------------------------------------
__global__ void finalize_kernel(const float* __restrict__ acc, float* __restrict__ out) {
    if (threadIdx.x == 0 && blockIdx.x == 0) {
        const float occ = acc[0] / (float)NR;
        const float cnt = fmaxf(acc[1], 1.0f);
        const float mdn = -acc[2] / (cnt * (float)TOPK);
        const float il  =  acc[3] / (cnt * (float)TOPK * (float)KMIX);
        out[0] = 0.2f * occ + 1.0f * mdn + 0.1f * il;
    }
}

// ---------------------------------------------------------------------------
extern "C" void kernel_launch(void* const* d_in, const int* in_sizes, int n_in,
                              void* d_out, int out_size, void* d_ws, size_t ws_size,
                              hipStream_t stream) {
    const float* mu_off  = (const float*)d_in[0];
    const float* log_sig = (const float*)d_in[1];
    const float* mu_int  = (const float*)d_in[2];
    const float* occ     = (const float*)d_in[3];
    const float* mix     = (const float*)d_in[4];
    const float* lfeat   = (const float*)d_in[5];
    const int*   ridx    = (const int*)d_in[6];
    const int*   lidx    = (const int*)d_in[7];
    float* out = (float*)d_out;

    char* ws = (char*)d_ws;
    float* acc    = (float*)ws;                                  // 4 floats
    int*   match  = (int*)(ws + 256);                            // NR ints
    int*   nn     = (int*)(ws + 256 + NR * sizeof(int));         // NR*TOPK ints

    init_acc_kernel<<<1, 32, 0, stream>>>(acc);
    match_topk_kernel<<<NR / 16, NTHREADS, 0, stream>>>(ridx, lidx, match, nn);
    loss_kernel<<<(NR * TOPK + 255) / 256, 256, 0, stream>>>(
        mu_off, log_sig, mu_int, occ, mix, lfeat, ridx, lidx, match, nn, acc);
    finalize_kernel<<<1, 32, 0, stream>>>(acc, out);
}